// AttentionSE3_67405216743683
// MI455X (gfx1250) — compile-verified
//
#include <hip/hip_runtime.h>
#include <hip/hip_bf16.h>
#include <math.h>
#include <limits.h>

// Problem constants (match reference)
#define NNODES   20000
#define NEDGES   640000
#define NHEADS   8
// features per head = 16, total features = 128, value row = 64*3 = 192 floats

typedef __attribute__((ext_vector_type(2))) float v2f;
typedef __attribute__((ext_vector_type(4))) float v4f;
typedef __attribute__((ext_vector_type(8))) float v8f;

// ---------------------------------------------------------------------------
// Ordered-int encoding: key(f) is monotonic in f over all ordered floats, so
// float segment-max == integer atomicMax on keys (single branchless atomic).
// ---------------------------------------------------------------------------
__device__ __forceinline__ int floatToOrderedInt(float f) {
    int i = __float_as_int(f);
    return i ^ ((i >> 31) & 0x7FFFFFFF);
}
__device__ __forceinline__ float orderedIntToFloat(int k) {
    return __int_as_float(k ^ ((k >> 31) & 0x7FFFFFFF));
}

// ---------------------------------------------------------------------------
// Pass 0: init m = INT_MIN (== -inf key), s = 0, zero feat output region
// ---------------------------------------------------------------------------
__global__ void init_kernel(int* __restrict__ m, float* __restrict__ s,
                            float* __restrict__ out, int nm, int nout) {
    int i = blockIdx.x * blockDim.x + threadIdx.x;
    if (i < nm) {
        m[i] = INT_MIN;
        s[i] = 0.0f;
    }
    if (i < nout) out[i] = 0.0f;
}

// ---------------------------------------------------------------------------
// Pass 1: prelogits via WMMA. One wave32 per 16-edge tile; loop 8 heads.
// For each head: C(16x16) = K_tile(16x16) * Q_tile^T(16x16) accumulated with
// four V_WMMA_F32_16X16X4_F32 steps; diagonal = per-edge logits.
//
// B operand staging (branch-free): for (node,h) the 16 q-features are
//   dd with dd%4==0  -> query_0[node*32 + h*4 + dd/4]       (4 floats, 16B aligned)
//   dd with dd%4>=1  -> query_1[(node*32+h*4)*3 + ...]      (12 floats, 48B -> 16B aligned)
// Lane-half lh needs K = {4kk+2lh, 4kk+2lh+1}:
//   lh=0: b = { q0v[kk],     q1v[3kk]   }
//   lh=1: b = { q1v[3kk+1],  q1v[3kk+2] }
//
// Diagonal extraction (1 ds_permute per head):
//   element (m,m): m<8  -> lane m,    VGPR c[m]
//                  m>=8 -> lane m+16, VGPR c[m-8]
//   Each source lane selects c[lane&7], then forward-permute routes
//   lane r -> lane r, lane 24+r -> lane 8+r (junk lanes -> lanes 16..31).
//
// All 8 segment-max atomics + prelogit stores deferred to one lane<16 block.
// ---------------------------------------------------------------------------
__global__ void logits_kernel(const float* __restrict__ key,
                              const float* __restrict__ q0,
                              const float* __restrict__ q1,
                              const int*  __restrict__ dst,
                              float* __restrict__ pre,   // (E,8)
                              int*   __restrict__ m) {   // (N,8) ordered-int keys
    const int ntiles = NEDGES / 16;
    int wave = (blockIdx.x * blockDim.x + threadIdx.x) >> 5;
    int lane = threadIdx.x & 31;
    if (wave >= ntiles) return;            // uniform per wave

    const int eBase = wave * 16;
    const int m16   = lane & 15;           // A: M row / B: N col
    const int lh    = lane >> 4;           // lane half selects K pair
    const int node  = dst[eBase + m16];    // gather once per tile
    const float* krow = key + (size_t)(eBase + m16) * 128;

    // Forward-permute routing (lane-only, hoisted):
    //   src 0..7   -> dst lane        (diag of edge lane)
    //   src 24..31 -> dst lane-16     (diag of edge lane-16)
    //   src 8..15  -> dst lane+16     (junk, don't care)
    //   src 16..23 -> dst lane        (junk, don't care)
    int dstLane = (lane < 8)  ? lane
                : (lane < 16) ? (lane + 16)
                : (lane < 24) ? lane
                              : (lane - 16);
    const int pidx = dstLane << 2;         // ds_permute index is in bytes

    float dv[NHEADS];                      // per-edge prelogits (lanes 0..15)

#pragma unroll
    for (int h = 0; h < NHEADS; ++h) {
        // Stage the full 16-feature q row for (node, h): all aligned v4f loads
        float q0v[4], q1v[12];
        *(v4f*)q0v       = *(const v4f*)(q0 + node * 32 + h * 4);
        const float* p1  = q1 + (size_t)(node * 32 + h * 4) * 3;
        *(v4f*)(q1v + 0) = *(const v4f*)(p1 + 0);
        *(v4f*)(q1v + 4) = *(const v4f*)(p1 + 4);
        *(v4f*)(q1v + 8) = *(const v4f*)(p1 + 8);

        v8f c = {};
#pragma unroll
        for (int kk = 0; kk < 4; ++kk) {
            // A-matrix 16x4 f32: lane m16 row, K = {kb, kb+1} (8B aligned pair)
            v2f a = *(const v2f*)(krow + h * 16 + kk * 4 + lh * 2);
            // B-matrix 4x16 f32: register selects, no control flow
            v2f b;
            b.x = lh ? q1v[3 * kk + 1] : q0v[kk];
            b.y = lh ? q1v[3 * kk + 2] : q1v[3 * kk];
            c = __builtin_amdgcn_wmma_f32_16x16x4_f32(
                    false, a, false, b, (short)0, c, false, false);
        }

        // Select this lane's diagonal-bearing register, then route in one op
        float send = c[0];
#pragma unroll
        for (int r = 1; r < 8; ++r) {
            if ((lane & 7) == r) send = c[r];
        }
        float diag = __int_as_float(
            __builtin_amdgcn_ds_permute(pidx, __float_as_int(send)));

        dv[h] = diag * 0.08838834764831845f;         // 1/sqrt(128)
    }

    // Single predicated tail: 8 branchless max-atomics + coalesced b128 stores
    if (lane < 16) {
        int* mrow = m + node * NHEADS;
#pragma unroll
        for (int h = 0; h < NHEADS; ++h) {
            atomicMax(mrow + h, floatToOrderedInt(dv[h]));
        }
        v4f* prow = (v4f*)(pre + (size_t)(eBase + lane) * NHEADS);
        v4f lo = { dv[0], dv[1], dv[2], dv[3] };
        v4f hi = { dv[4], dv[5], dv[6], dv[7] };
        prow[0] = lo;
        prow[1] = hi;
    }
}

// ---------------------------------------------------------------------------
// Pass 2: ex = exp(pre - m[dst]); segment-sum into s via f32 atomics
// ---------------------------------------------------------------------------
__global__ void sum_kernel(const float* __restrict__ pre,
                           const int*  __restrict__ dst,
                           const int*  __restrict__ m,
                           float* __restrict__ s) {
    int i = blockIdx.x * blockDim.x + threadIdx.x;   // over E*H
    if (i >= NEDGES * NHEADS) return;
    int e = i >> 3;
    int h = i & 7;
    int node = dst[e];
    float mv = orderedIntToFloat(m[node * NHEADS + h]);
    float ex = __expf(pre[i] - mv);
    atomicAdd(&s[node * NHEADS + h], ex);
}

// ---------------------------------------------------------------------------
// Pass 3: weighted scatter-add of value into out. Only the 128 surviving
// elements per edge are accumulated:
//   c <  32 : d = 0 only        -> out_0[node*32 + c]
//   c >= 32 : d in {0,1,2}      -> out_1[(node*32 + c-32)*3 + d]
// 128 threads per edge; w recomputed from pre/m/s (all L2-resident gathers).
// ---------------------------------------------------------------------------
__global__ void accum_kernel(const float* __restrict__ value,   // (E,64,3)
                             const float* __restrict__ pre,     // (E,8)
                             const int*  __restrict__ dst,
                             const int*  __restrict__ m,
                             const float* __restrict__ s,
                             float* __restrict__ out) {
    int i = blockIdx.x * blockDim.x + threadIdx.x;   // over E*128
    if (i >= NEDGES * 128) return;
    int e = i >> 7;
    int t = i & 127;
    int node = dst[e];

    int c, d, outIdx;
    if (t < 32) {
        c = t; d = 0;
        outIdx = node * 32 + c;                                  // out_0 region
    } else {
        int u = t - 32;
        int cc = u / 3;
        d = u - cc * 3;
        c = 32 + cc;
        outIdx = NNODES * 32 + (node * 32 + cc) * 3 + d;         // out_1 region
    }
    int h = c >> 3;
    float mv = orderedIntToFloat(m[node * NHEADS + h]);
    float w = __expf(pre[e * NHEADS + h] - mv) / s[node * NHEADS + h];
    float val = value[(size_t)e * 192 + c * 3 + d];
    atomicAdd(&out[outIdx], w * val);
}

// ---------------------------------------------------------------------------
extern "C" void kernel_launch(void* const* d_in, const int* in_sizes, int n_in,
                              void* d_out, int out_size, void* d_ws, size_t ws_size,
                              hipStream_t stream) {
    const float* value = (const float*)d_in[0];   // E*64*3
    const float* key   = (const float*)d_in[1];   // E*128
    const float* q0    = (const float*)d_in[2];   // N*32
    const float* q1    = (const float*)d_in[3];   // N*32*3
    const int*   dst   = (const int*)d_in[4];     // E

    float* out  = (float*)d_out;
    float* pre  = out + NNODES * 32 + NNODES * 96;   // prelogits region (E*8)

    int*   m = (int*)d_ws;                           // N*8 ordered-int keys
    float* s = (float*)(m + NNODES * NHEADS);        // N*8

    const int B = 256;

    // Pass 0: init m/s and zero the feat output region (N*128 floats)
    {
        int n = NNODES * 128;
        init_kernel<<<(n + B - 1) / B, B, 0, stream>>>(m, s, out,
                                                       NNODES * NHEADS, n);
    }
    // Pass 1: WMMA logits + segment max
    {
        int waves = NEDGES / 16;                     // 40000 wave32 tiles
        int threads = waves * 32;
        logits_kernel<<<(threads + B - 1) / B, B, 0, stream>>>(key, q0, q1, dst,
                                                               pre, m);
    }
    // Pass 2: exp + segment sum
    {
        int n = NEDGES * NHEADS;
        sum_kernel<<<(n + B - 1) / B, B, 0, stream>>>(pre, dst, m, s);
    }
    // Pass 3: weighted scatter-add into outputs
    {
        int n = NEDGES * 128;
        accum_kernel<<<(n + B - 1) / B, B, 0, stream>>>(value, pre, dst, m, s,
                                                        out);
    }
}